// AtomsNetwork_27058293965312
// MI455X (gfx1250) — compile-verified
//
#include <hip/hip_runtime.h>
#include <hip/hip_bf16.h>

// ---------------------------------------------------------------------------
// AtomsNetwork on gfx1250 (MI455X), fp32 WMMA (V_WMMA_F32_16X16X4_F32) path.
// ---------------------------------------------------------------------------

typedef float v2f __attribute__((ext_vector_type(2)));
typedef float v8f __attribute__((ext_vector_type(8)));

#define N_ATOMS 16000
#define KNBR    10
#define N_RES   400
#define APR     40          // atoms per residue (sorted, contiguous)
#define ATOM_CAT 12
#define BERT_DIM 1024
#define FEAT    128
#define DF1     256
#define DF2     64

__device__ __forceinline__ v8f wmma_f32_k4(v2f a, v2f b, v8f c) {
    // D = A(16x4 f32) * B(4x16 f32) + C(16x16 f32)
    return __builtin_amdgcn_wmma_f32_16x16x4_f32(
        /*neg_a=*/false, a, /*neg_b=*/false, b,
        /*c_mod=*/(short)0, c, /*reuse_a=*/false, /*reuse_b=*/false);
}

// ---------------------------------------------------------------------------
// fp32 WMMA GEMM:  C[M,N] = A[M,K] * B[K,N]   (row-major, N%16==0, K%4==0,
// M % (16*MB) == 0).  One wave computes a (16*MB)x16 C strip: the B fragment
// is loaded once per K-step and reused across MB WMMAs (A-streamed).
// Pure pointer-increment addressing in the inner loop.
// ---------------------------------------------------------------------------
template<int MB>
__global__ __launch_bounds__(256)
void gemm_f32_wmma(const float* __restrict__ A, const float* __restrict__ B,
                   float* __restrict__ C, int M, int N, int K) {
    const int lane = threadIdx.x & 31;
    const int wave = threadIdx.x >> 5;
    const int ntn  = N >> 4;
    const int ntm  = M / (16 * MB);
    const int tile = blockIdx.x * 8 + wave;
    if (tile >= ntm * ntn) return;               // wave-uniform: EXEC stays full
    const int tm = tile / ntn;
    const int tn = tile - tm * ntn;

    const int ln   = lane & 15;
    const int half = lane >> 4;
    const int koff = half * 2;

    const float* ap[MB];
#pragma unroll
    for (int m = 0; m < MB; ++m)
        ap[m] = A + (size_t)(tm * 16 * MB + m * 16 + ln) * (size_t)K + koff;
    const float* bp = B + (size_t)koff * N + tn * 16 + ln;
    const size_t bstep = (size_t)4 * N;

    v8f acc[MB];
#pragma unroll
    for (int m = 0; m < MB; ++m) acc[m] = (v8f){0.f,0.f,0.f,0.f,0.f,0.f,0.f,0.f};

#pragma unroll 2
    for (int kb = 0; kb < K; kb += 4) {
        v2f b;
        b.x = bp[0];
        b.y = bp[N];
        bp += bstep;
#pragma unroll
        for (int m = 0; m < MB; ++m) {
            v2f a = *(const v2f*)ap[m];          // 8B-aligned b64 load
            ap[m] += 4;
            acc[m] = wmma_f32_k4(a, b, acc[m]);  // B-frag reused MB times
        }
    }

    const int col = tn * 16 + ln;
#pragma unroll
    for (int m = 0; m < MB; ++m) {
        // C/D layout: VGPR v, lane -> row = base + v + 8*half, col = tn*16+ln
        float* cp = C + (size_t)(tm * 16 * MB + m * 16 + half * 8) * N + col;
#pragma unroll
        for (int v = 0; v < 8; ++v) cp[(size_t)v * N] = acc[m][v];
    }
}

// ---------------------------------------------------------------------------
// combine: z = relu(base0 [+ base1] + neigh_mean(s, sn) + neigh_mean(d, dn))
// grid = N_ATOMS blocks, block = FEAT threads (thread = feature).
// ---------------------------------------------------------------------------
__global__ __launch_bounds__(FEAT)
void combine_kernel(const float* __restrict__ base0, const float* __restrict__ base1,
                    const float* __restrict__ s, const float* __restrict__ d,
                    const int* __restrict__ sn, const int* __restrict__ dn,
                    float* __restrict__ z) {
    const int n = blockIdx.x;
    const int f = threadIdx.x;
    float acc = base0[(size_t)n * FEAT + f];
    if (base1) acc += base1[(size_t)n * FEAT + f];

    float ss = 0.f; int cs = 0;
#pragma unroll
    for (int k = 0; k < KNBR; ++k) {
        const int idx = sn[n * KNBR + k];
        if (idx > -1) { ss += s[(size_t)idx * FEAT + f]; ++cs; }
    }
    acc += ss / (float)(cs > 0 ? cs : 1);

    float ds = 0.f; int cd = 0;
#pragma unroll
    for (int k = 0; k < KNBR; ++k) {
        const int idx = dn[n * KNBR + k];
        if (idx > -1) { ds += d[(size_t)idx * FEAT + f]; ++cd; }
    }
    acc += ds / (float)(cd > 0 ? cd : 1);

    z[(size_t)n * FEAT + f] = fmaxf(acc, 0.f);
}

// ---------------------------------------------------------------------------
// Residue mean pooling (sorted contiguous residue ids, APR atoms each).
// ---------------------------------------------------------------------------
__global__ __launch_bounds__(FEAT)
void res_mean_kernel(const float* __restrict__ z, float* __restrict__ r) {
    const int rb = blockIdx.x;
    const int f  = threadIdx.x;
    float s = 0.f;
    const float* zp = z + (size_t)rb * APR * FEAT + f;
#pragma unroll 8
    for (int a = 0; a < APR; ++a) s += zp[(size_t)a * FEAT];
    r[(size_t)rb * FEAT + f] = s * (1.0f / (float)APR);
}

// ---------------------------------------------------------------------------
// Fused pair head.  For pair (i, j):
//   h1 = relu(A1[i,:] + B1[j,:] + bf1)            [256]
//   h2 = relu(h1 @ Wf2 + bf2)                     [64]
//   out[i*400+j] = h2 @ Wf3 + bf3
// One wave handles 16 pairs (fixed i, 16 consecutive j) x all 64 h2 outputs
// via 4 WMMA accumulators over K=256.  Wf2 is staged TRANSPOSED in LDS with a
// 258-float pad so each B fragment is one conflict-free aligned ds_load_b64;
// A1[i]+bf1 is pre-summed into LDS.
// grid = (N_RES, 4), block = 256 (8 waves; j-tiles 0..24 used).
// ---------------------------------------------------------------------------
#define WT_STRIDE 258
__global__ __launch_bounds__(256)
void head_pairs_kernel(const float* __restrict__ A1, const float* __restrict__ B1,
                       const float* __restrict__ bf1, const float* __restrict__ Wf2,
                       const float* __restrict__ bf2, const float* __restrict__ Wf3,
                       const float* __restrict__ bf3, float* __restrict__ out) {
    __shared__ float sWf2T[DF2 * WT_STRIDE];   // Wf2 transposed, padded (64.5 KB)
    __shared__ float sAb[DF1];                 // A1[i,:] + bf1
    __shared__ float sbf2[DF2];
    __shared__ float sWf3[DF2];

    const int i = blockIdx.x;
    const int t = threadIdx.x;
    for (int idx = t; idx < DF1 * DF2; idx += 256) {
        const int k = idx >> 6;                // row of Wf2 (K index)
        const int n = idx & 63;                // col of Wf2 (N index)
        sWf2T[n * WT_STRIDE + k] = Wf2[idx];   // pad 258 -> conflict-free
    }
    sAb[t] = A1[(size_t)i * DF1 + t] + bf1[t];
    if (t < DF2) { sbf2[t] = bf2[t]; sWf3[t] = Wf3[t]; }
    __syncthreads();

    const int wave = t >> 5;
    const int lane = t & 31;
    const int jt = blockIdx.y * 8 + wave;
    if (jt >= (N_RES / 16)) return;            // wave-uniform skip
    const int j0   = jt * 16;
    const int ln   = lane & 15;
    const int half = lane >> 4;
    const int koff = half * 2;

    v8f acc[4];
#pragma unroll
    for (int q = 0; q < 4; ++q) acc[q] = (v8f){0.f,0.f,0.f,0.f,0.f,0.f,0.f,0.f};

    const float* B1row = B1 + (size_t)(j0 + ln) * DF1 + koff;
    for (int kb = 0; kb < DF1; kb += 4) {
        const int k0 = kb + koff;
        v2f bv = *(const v2f*)(B1row + kb);            // global b64, 8B aligned
        v2f av = *(const v2f*)(&sAb[k0]);              // ds b64, 8B aligned
        v2f a;
        a.x = fmaxf(av.x + bv.x, 0.f);
        a.y = fmaxf(av.y + bv.y, 0.f);
#pragma unroll
        for (int q = 0; q < 4; ++q) {
            v2f b = *(const v2f*)(&sWf2T[(q * 16 + ln) * WT_STRIDE + k0]); // ds b64
            acc[q] = wmma_f32_k4(a, b, acc[q]);        // A-frag reused 4x
        }
    }

    // Epilogue: relu(h2 + bf2) dot Wf3, reduced across the 16 lanes of a half.
    float part[8];
#pragma unroll
    for (int v = 0; v < 8; ++v) part[v] = 0.f;
#pragma unroll
    for (int q = 0; q < 4; ++q) {
        const int n = q * 16 + ln;
        const float w  = sWf3[n];
        const float bb = sbf2[n];
#pragma unroll
        for (int v = 0; v < 8; ++v)
            part[v] += fmaxf(acc[q][v] + bb, 0.f) * w;
    }
#pragma unroll
    for (int off = 8; off >= 1; off >>= 1) {
#pragma unroll
        for (int v = 0; v < 8; ++v) part[v] += __shfl_xor(part[v], off, 32);
    }
    if (ln == 0) {
        const float b3 = bf3[0];
        const int jm = j0 + half * 8;          // half 0 -> pairs 0..7, half 1 -> 8..15
#pragma unroll
        for (int v = 0; v < 8; ++v)
            out[(size_t)i * N_RES + jm + v] = part[v] + b3;
    }
}

// ---------------------------------------------------------------------------
// Host orchestration.
// ---------------------------------------------------------------------------
template<int MB>
static inline void launch_gemm(const float* A, const float* B, float* C,
                               int M, int N, int K, hipStream_t stream) {
    const int tiles = (M / (16 * MB)) * (N >> 4);
    const int grid  = (tiles + 7) / 8;
    gemm_f32_wmma<MB><<<grid, 256, 0, stream>>>(A, B, C, M, N, K);
}

extern "C" void kernel_launch(void* const* d_in, const int* in_sizes, int n_in,
                              void* d_out, int out_size, void* d_ws, size_t ws_size,
                              hipStream_t stream) {
    (void)in_sizes; (void)n_in; (void)out_size; (void)ws_size;

    const float* atoms[2]    = { (const float*)d_in[0], (const float*)d_in[5] };
    const float* residues[2] = { (const float*)d_in[1], (const float*)d_in[6] };
    const int*   same[2]     = { (const int*)d_in[2],   (const int*)d_in[7]   };
    const int*   diff[2]     = { (const int*)d_in[3],   (const int*)d_in[8]   };
    const float* Wv   = (const float*)d_in[10];
    const float* Wr   = (const float*)d_in[11];
    const float* Wsr1 = (const float*)d_in[12];
    const float* Wdr1 = (const float*)d_in[13];
    const float* Wsv  = (const float*)d_in[14];
    const float* Wsr2 = (const float*)d_in[15];
    const float* Wdr2 = (const float*)d_in[16];
    const float* Wf1  = (const float*)d_in[17];
    const float* bf1  = (const float*)d_in[18];
    const float* Wf2  = (const float*)d_in[19];
    const float* bf2  = (const float*)d_in[20];
    const float* Wf3  = (const float*)d_in[21];
    const float* bf3  = (const float*)d_in[22];
    float* out = (float*)d_out;

    // Workspace carve-up (floats). 5 atom-feature buffers reused across stages.
    const size_t SZ = (size_t)N_ATOMS * FEAT;       // 2.048 M floats
    float* ws   = (float*)d_ws;
    float* buf0 = ws;                               // pv  / t0
    float* buf1 = ws + 1 * SZ;                      // pr  / t1
    float* buf2 = ws + 2 * SZ;                      // s1  / t2
    float* buf3 = ws + 3 * SZ;                      // d1  / z2
    float* buf4 = ws + 4 * SZ;                      // z1
    float* rres[2] = { ws + 5 * SZ, ws + 5 * SZ + (size_t)N_RES * FEAT };
    float* A1 = ws + 5 * SZ + 2 * (size_t)N_RES * FEAT;
    float* B1 = A1 + (size_t)N_RES * DF1;

    for (int p = 0; p < 2; ++p) {
        // --- GNN first layer ---
        launch_gemm<4>(atoms[p],    Wv,   buf0, N_ATOMS, FEAT, ATOM_CAT, stream);
        launch_gemm<4>(residues[p], Wr,   buf1, N_ATOMS, FEAT, BERT_DIM, stream);
        launch_gemm<4>(atoms[p],    Wsr1, buf2, N_ATOMS, FEAT, ATOM_CAT, stream);
        launch_gemm<4>(atoms[p],    Wdr1, buf3, N_ATOMS, FEAT, ATOM_CAT, stream);
        combine_kernel<<<N_ATOMS, FEAT, 0, stream>>>(
            buf0, buf1, buf2, buf3, same[p], diff[p], buf4);      // z1 -> buf4

        // --- GNN second layer ---
        launch_gemm<4>(buf4, Wsv,  buf0, N_ATOMS, FEAT, FEAT, stream);
        launch_gemm<4>(buf4, Wsr2, buf1, N_ATOMS, FEAT, FEAT, stream);
        launch_gemm<4>(buf4, Wdr2, buf2, N_ATOMS, FEAT, FEAT, stream);
        combine_kernel<<<N_ATOMS, FEAT, 0, stream>>>(
            buf0, nullptr, buf1, buf2, same[p], diff[p], buf3);   // z2 -> buf3

        // --- residue mean pooling ---
        res_mean_kernel<<<N_RES, FEAT, 0, stream>>>(buf3, rres[p]);
    }

    // Head layer-1 factorization: x@Wf1 = r1@Wf1[:128] (+) r2@Wf1[128:]
    launch_gemm<1>(rres[0], Wf1,                         A1, N_RES, DF1, FEAT, stream);
    launch_gemm<1>(rres[1], Wf1 + (size_t)FEAT * DF1,    B1, N_RES, DF1, FEAT, stream);

    // Fused pair head over all 160k pairs.
    head_pairs_kernel<<<dim3(N_RES, 4), 256, 0, stream>>>(
        A1, B1, bf1, Wf2, bf2, Wf3, bf3, out);
}